// DifferentiableStack_24902220382568
// MI455X (gfx1250) — compile-verified
//
#include <hip/hip_runtime.h>
#include <cstdint>

#define B_DIM   1024
#define L_DIM   1024
#define D_DIM   128
#define THREADS 256   // 8 waves (wave32)
#define DEPTH   8     // async pipeline depth (rows in flight per wave)

// out[b,d] = sum_t x[b,t,d] * p[b,t] * prod_{u>t} (1 - o[b,u])
__global__ __launch_bounds__(THREADS)
void diff_stack_kernel(const float* __restrict__ x,
                       const float* __restrict__ push_gate,
                       const float* __restrict__ pop_gate,
                       float* __restrict__ out)
{
    __shared__ float4 wlds4[L_DIM / 4];        //  4 KB: weights w[t]
    __shared__ float4 xbuf4[DEPTH][8][32];     // 32 KB: 8-deep ring of x rows, per wave
    __shared__ float4 red4[8][32];             //  4 KB: cross-wave reduction
    __shared__ float  wtot[8];

    const int tid  = threadIdx.x;
    const int lane = tid & 31;
    const int wid  = tid >> 5;
    const int b    = blockIdx.x;

    // ---------------- Phase 1: suffix-product weights --------------------
    // thread owns t = 4*tid .. 4*tid+3
    const float4 p4 = ((const float4*)(push_gate + (size_t)b * L_DIM))[tid];
    const float4 o4 = ((const float4*)(pop_gate  + (size_t)b * L_DIM))[tid];
    float4 q4;
    q4.x = 1.0f - o4.x; q4.y = 1.0f - o4.y;
    q4.z = 1.0f - o4.z; q4.w = 1.0f - o4.w;

    float P = q4.x * q4.y * q4.z * q4.w;

    // wave32 inclusive suffix product: incl[i] = prod_{j>=i} P[j]
    float incl = P;
    #pragma unroll
    for (int off = 1; off < 32; off <<= 1) {
        float n = __shfl_down(incl, off, 32);
        if (lane + off < 32) incl *= n;
    }
    if (lane == 0) wtot[wid] = incl;            // whole-wave product
    float nx   = __shfl_down(incl, 1, 32);
    float excl = (lane == 31) ? 1.0f : nx;      // prod over lanes j>lane
    __syncthreads();

    float suffW = 1.0f;                          // prod over waves after mine
    #pragma unroll
    for (int wv = 0; wv < 8; ++wv)
        if (wv > wid) suffW *= wtot[wv];

    float r = excl * suffW;                      // prod of q over all t > 4*tid+3
    float4 wv4;
    wv4.w = p4.w * r; r *= q4.w;
    wv4.z = p4.z * r; r *= q4.z;
    wv4.y = p4.y * r; r *= q4.y;
    wv4.x = p4.x * r;
    wlds4[tid] = wv4;
    __syncthreads();

    // ---------------- Phase 2: out[b,:] = sum_t w[t] * x[b,t,:] ----------
    // Wave `wid` handles rows t = wid, wid+8, ..., wid+1016 (128 rows),
    // streamed through an 8-deep per-wave LDS ring via CDNA5 async copies
    // (non-temporal: 512 MB single-pass stream must not thrash 192 MB L2).
    const float* wl = (const float*)wlds4;
    const float* xb = x + (size_t)b * (L_DIM * D_DIM);

    // compiler-visible init so LDS reads can't be folded away
    #pragma unroll
    for (int k = 0; k < DEPTH; ++k)
        xbuf4[k][wid][lane] = make_float4(0.f, 0.f, 0.f, 0.f);
    asm volatile("s_wait_dscnt 0" ::: "memory");

    uint32_t lA[DEPTH];
    #pragma unroll
    for (int k = 0; k < DEPTH; ++k)
        lA[k] = (uint32_t)(uintptr_t)&xbuf4[k][wid][lane];

    const uint64_t g0 = (uint64_t)(uintptr_t)(xb + (size_t)wid * D_DIM)
                      + (uint64_t)lane * 16u;
    const uint64_t rowStride = 8u * D_DIM * 4u;   // 4096 B between this wave's rows

    // prologue: fill the ring (DEPTH rows in flight per wave)
    #pragma unroll
    for (int k = 0; k < DEPTH; ++k) {
        const uint64_t gk = g0 + (uint64_t)k * rowStride;
        asm volatile("global_load_async_to_lds_b128 %0, %1, off th:TH_LOAD_NT"
                     :: "v"(lA[k]), "v"(gk) : "memory");
    }

    float4 acc = make_float4(0.f, 0.f, 0.f, 0.f);

    #pragma unroll 8
    for (int i = 0; i < 128 - DEPTH; ++i) {
        // in-order completion: <=DEPTH-1 outstanding => row i has landed
        asm volatile("s_wait_asynccnt 7" ::: "memory");
        const float  wt = wl[wid + 8 * i];        // LDS broadcast read
        const float4 v  = xbuf4[i & (DEPTH - 1)][wid][lane];
        acc.x += wt * v.x;
        acc.y += wt * v.y;
        acc.z += wt * v.z;
        acc.w += wt * v.w;
        // ds reads must retire before the async engine overwrites this slot
        asm volatile("s_wait_dscnt 0" ::: "memory");
        const uint64_t gn = g0 + (uint64_t)(i + DEPTH) * rowStride;
        asm volatile("global_load_async_to_lds_b128 %0, %1, off th:TH_LOAD_NT"
                     :: "v"(lA[i & (DEPTH - 1)]), "v"(gn) : "memory");
    }

    // drain the ring
    asm volatile("s_wait_asynccnt 0" ::: "memory");
    #pragma unroll
    for (int i = 128 - DEPTH; i < 128; ++i) {
        const float  wt = wl[wid + 8 * i];
        const float4 v  = xbuf4[i & (DEPTH - 1)][wid][lane];
        acc.x += wt * v.x;
        acc.y += wt * v.y;
        acc.z += wt * v.z;
        acc.w += wt * v.w;
    }

    red4[wid][lane] = acc;
    __syncthreads();

    // ---------------- Phase 3: cross-wave reduce, write out --------------
    if (tid < 32) {
        float4 s = make_float4(0.f, 0.f, 0.f, 0.f);
        #pragma unroll
        for (int wv = 0; wv < 8; ++wv) {
            const float4 vv = red4[wv][tid];
            s.x += vv.x; s.y += vv.y; s.z += vv.z; s.w += vv.w;
        }
        ((float4*)(out + (size_t)b * D_DIM))[tid] = s;
    }
}

extern "C" void kernel_launch(void* const* d_in, const int* in_sizes, int n_in,
                              void* d_out, int out_size, void* d_ws, size_t ws_size,
                              hipStream_t stream) {
    const float* x  = (const float*)d_in[0];   // (B, L, D) f32
    const float* pg = (const float*)d_in[1];   // (B, L, 1) f32
    const float* og = (const float*)d_in[2];   // (B, L, 1) f32
    float* outp     = (float*)d_out;           // (B, D) f32

    diff_stack_kernel<<<dim3(B_DIM), dim3(THREADS), 0, stream>>>(x, pg, og, outp);
}